// TCN_GCN_unit_3977139716232
// MI455X (gfx1250) — compile-verified
//
#include <hip/hip_runtime.h>
#include <hip/hip_bf16.h>

// ---------------------------------------------------------------------------
// ST-GCN TCN_GCN_unit for MI455X (gfx1250), fp32 WMMA (16x16x4) path.
// N=64, C_in=C_out=64, T=300, V=25, K=3, KT=9.
// ---------------------------------------------------------------------------

typedef float v2f __attribute__((ext_vector_type(2)));
typedef float v8f __attribute__((ext_vector_type(8)));

#define N_BATCH 64
#define C_DIM   64
#define T_DIM   300
#define V_DIM   25
#define K_SETS  3
#define KT_DIM  9
#define R_DIM   (C_DIM * V_DIM)        // 1600 : fused (c,w) / (ci,v) dimension
#define KTCN    (C_DIM * KT_DIM)       // 576  : im2col K for temporal conv
#define CNT_ELEMS (N_BATCH * T_DIM * V_DIM)  // 480000 per-channel BN count

__device__ __forceinline__ v8f wmma_f32(v2f a, v2f b, v8f c) {
  // D = A(16x4) x B(4x16) + C(16x16), fp32 matrix pipe.
  return __builtin_amdgcn_wmma_f32_16x16x4_f32(
      /*neg_a=*/false, a, /*neg_b=*/false, b,
      /*c_mod=*/(short)0, c, /*reuse_a=*/false, /*reuse_b=*/false);
}

// ---------------------------------------------------------------------------
// Kernel 0: Mbig[(c,w),(ci,v)] = sum_k Wg[k*64+c, ci] * A[k, v, w]
// ---------------------------------------------------------------------------
__global__ __launch_bounds__(256) void build_mbig(
    const float* __restrict__ wg, const float* __restrict__ A,
    float* __restrict__ Mbig) {
  int idx = blockIdx.x * 256 + threadIdx.x;
  if (idx >= R_DIM * R_DIM) return;
  int r = idx / R_DIM;         // (c, w)
  int s = idx % R_DIM;         // (ci, v)
  int c  = r / V_DIM, w = r % V_DIM;
  int ci = s / V_DIM, v = s % V_DIM;
  float acc = 0.f;
#pragma unroll
  for (int k = 0; k < K_SETS; ++k)
    acc += wg[(k * C_DIM + c) * C_DIM + ci] * A[(k * V_DIM + v) * V_DIM + w];
  Mbig[idx] = acc;
}

// ---------------------------------------------------------------------------
// Kernel 0b: bterm[(c,w)] = sum_k b_gcn[k*64+c] * sum_v A[k,v,w]; zero stats
// ---------------------------------------------------------------------------
__global__ __launch_bounds__(256) void build_bterm(
    const float* __restrict__ bg, const float* __restrict__ A,
    float* __restrict__ bterm, float* __restrict__ gstat) {
  int idx = blockIdx.x * 256 + threadIdx.x;
  if (idx < 128) gstat[idx] = 0.f;       // zero BN accumulators (sum, sumsq)
  if (idx >= R_DIM) return;
  int c = idx / V_DIM, w = idx % V_DIM;
  float acc = 0.f;
#pragma unroll
  for (int k = 0; k < K_SETS; ++k) {
    float cs = 0.f;
#pragma unroll
    for (int v = 0; v < V_DIM; ++v) cs += A[(k * V_DIM + v) * V_DIM + w];
    acc += bg[k * C_DIM + c] * cs;
  }
  bterm[idx] = acc;
}

// ---------------------------------------------------------------------------
// Kernel 1: fused GCN  Z[n, r, t] = Mbig[r, :] . Xv[n, :, t] + bterm[r]
//   M=1600, K=1600, N=300 (per batch). Block tile 128x64, 8 waves (4Mx2N),
//   each wave: 2x2 WMMA 16x16 tiles, fp32 WMMA with K-step 4.
// ---------------------------------------------------------------------------
__global__ __launch_bounds__(256) void gcn_gemm(
    const float* __restrict__ x, const float* __restrict__ Mbig,
    const float* __restrict__ bterm, float* __restrict__ Z) {
  __shared__ float Ms[128][17];   // padded: stride 17 -> bank-conflict-free A frags
  __shared__ float Xs[16][64];

  const int tid    = threadIdx.x;
  const int n      = blockIdx.z;
  const int m_base = blockIdx.y * 128;
  const int t_base = blockIdx.x * 64;
  const int lane   = tid & 31;
  const int wv     = tid >> 5;
  const int wm     = wv & 3;       // 0..3 -> M strip
  const int wn     = wv >> 2;      // 0..1 -> N strip
  const int h      = (lane >> 4) & 1;
  const int lr     = lane & 15;

  const v8f vzero = {0.f, 0.f, 0.f, 0.f, 0.f, 0.f, 0.f, 0.f};
  v8f acc[2][2];
#pragma unroll
  for (int mt = 0; mt < 2; ++mt)
#pragma unroll
    for (int nt = 0; nt < 2; ++nt) acc[mt][nt] = vzero;

  for (int kb = 0; kb < R_DIM; kb += 16) {
    // --- stage Mbig tile: 128 rows x 16 cols (2 float4 per thread) ---
#pragma unroll
    for (int j = 0; j < 2; ++j) {
      int fid = tid + j * 256;
      int row = fid >> 2;
      int c4  = (fid & 3) << 2;
      int gr  = m_base + row;
      float4 val = make_float4(0.f, 0.f, 0.f, 0.f);
      if (gr < R_DIM)
        val = *(const float4*)&Mbig[(size_t)gr * R_DIM + kb + c4];
      Ms[row][c4 + 0] = val.x;
      Ms[row][c4 + 1] = val.y;
      Ms[row][c4 + 2] = val.z;
      Ms[row][c4 + 3] = val.w;
    }
    // --- stage X tile: 16 (ci,v)-rows x 64 t-cols ---
#pragma unroll
    for (int j = 0; j < 4; ++j) {
      int eid = tid + j * 256;
      int sl  = eid >> 6;          // 0..15
      int tl  = eid & 63;
      int s   = kb + sl;
      int ci  = s / V_DIM, v = s % V_DIM;
      int t   = t_base + tl;
      float val = 0.f;
      if (t < T_DIM)
        val = x[((size_t)(n * C_DIM + ci) * T_DIM + t) * V_DIM + v];
      Xs[sl][tl] = val;
    }
    // prefetch next Mbig tile into cache hierarchy
    if (kb + 16 < R_DIM) {
      int gr = m_base + (tid >> 1);
      if (gr < R_DIM)
        __builtin_prefetch(&Mbig[(size_t)gr * R_DIM + kb + 16 + ((tid & 1) << 3)], 0, 1);
    }
    __syncthreads();

#pragma unroll
    for (int kk = 0; kk < 16; kk += 4) {
      v2f a[2], b[2];
#pragma unroll
      for (int mt = 0; mt < 2; ++mt) {
        int row = wm * 32 + mt * 16 + lr;
        a[mt].x = Ms[row][kk + 2 * h + 0];
        a[mt].y = Ms[row][kk + 2 * h + 1];
      }
#pragma unroll
      for (int nt = 0; nt < 2; ++nt) {
        int col = wn * 32 + nt * 16 + lr;
        b[nt].x = Xs[kk + 2 * h + 0][col];
        b[nt].y = Xs[kk + 2 * h + 1][col];
      }
#pragma unroll
      for (int mt = 0; mt < 2; ++mt)
#pragma unroll
        for (int nt = 0; nt < 2; ++nt)
          acc[mt][nt] = wmma_f32(a[mt], b[nt], acc[mt][nt]);
    }
    __syncthreads();
  }

  // --- epilogue: + bterm, store Z[n][r][t] ---
#pragma unroll
  for (int mt = 0; mt < 2; ++mt)
#pragma unroll
    for (int nt = 0; nt < 2; ++nt)
#pragma unroll
      for (int i = 0; i < 8; ++i) {
        int r = m_base + wm * 32 + mt * 16 + i + 8 * h;
        int t = t_base + wn * 32 + nt * 16 + lr;
        if (r < R_DIM && t < T_DIM)
          Z[((size_t)n * R_DIM + r) * T_DIM + t] = acc[mt][nt][i] + bterm[r];
      }
}

// ---------------------------------------------------------------------------
// Kernel 2: temporal conv as implicit GEMM per (n, w):
//   Y[o,t] = Wt[o,(i,dt)] . Zs[(i,dt),t] + b_tcn[o];  M=64, K=576, N=300.
//   Full T row (+halo, zero padded to 320+8) staged in LDS. Fused BN-stat
//   accumulation (per-channel sum / sumsq) via LDS + global float atomics.
//   8 waves as 2(M) x 4(N); each wave: 2x5 WMMA tiles.
// ---------------------------------------------------------------------------
#define ZS_COLS 328   // 320 padded t + 8 halo reach
__global__ __launch_bounds__(256) void tcn_gemm(
    const float* __restrict__ Z, const float* __restrict__ w_tcn,
    const float* __restrict__ b_tcn, float* __restrict__ Y,
    float* __restrict__ gstat) {
  extern __shared__ float smem[];
  float* Zs    = smem;                        // [64][ZS_COLS]
  float* Ast   = smem + C_DIM * ZS_COLS;      // [16][64] transposed W tile
  float* sstat = Ast + 16 * 64;               // [128]

  const int tid  = threadIdx.x;
  const int w    = blockIdx.x;                // vertex 0..24
  const int n    = blockIdx.y;                // batch
  const int lane = tid & 31;
  const int wv   = tid >> 5;
  const int wm   = wv & 1;                    // 0..1 -> M strip (32 rows)
  const int wn   = wv >> 1;                   // 0..3 -> N strip (80 cols)
  const int h    = (lane >> 4) & 1;
  const int lr   = lane & 15;

  if (tid < 128) sstat[tid] = 0.f;

  // stage Z row block: Zs[i][j] = Z[n, i*25+w, j-4] (0 outside [0,300))
  for (int idx = tid; idx < C_DIM * ZS_COLS; idx += 256) {
    int i = idx / ZS_COLS;
    int j = idx % ZS_COLS;
    int t = j - 4;
    float val = 0.f;
    if (t >= 0 && t < T_DIM)
      val = Z[((size_t)n * R_DIM + i * V_DIM + w) * T_DIM + t];
    Zs[idx] = val;
  }

  const v8f vzero = {0.f, 0.f, 0.f, 0.f, 0.f, 0.f, 0.f, 0.f};
  v8f acc[2][5];
#pragma unroll
  for (int mt = 0; mt < 2; ++mt)
#pragma unroll
    for (int nt = 0; nt < 5; ++nt) acc[mt][nt] = vzero;

  for (int kb = 0; kb < KTCN; kb += 16) {
    __syncthreads();  // (first iter: Zs+sstat ready; later: Ast consumed)
    // stage transposed weight tile: Ast[kl][o] = w_tcn[o][kb+kl]
#pragma unroll
    for (int j = 0; j < 4; ++j) {
      int eid = tid + j * 256;
      int kl  = eid >> 6;
      int o   = eid & 63;
      Ast[kl * 64 + o] = w_tcn[(size_t)o * KTCN + kb + kl];
    }
    __syncthreads();

#pragma unroll
    for (int kk = 0; kk < 16; kk += 4) {
      v2f a[2];
#pragma unroll
      for (int mt = 0; mt < 2; ++mt) {
        int o = wm * 32 + mt * 16 + lr;
        a[mt].x = Ast[(kk + 2 * h + 0) * 64 + o];
        a[mt].y = Ast[(kk + 2 * h + 1) * 64 + o];
      }
      int ki0 = kb + kk + 2 * h;
      int i0 = ki0 / KT_DIM, d0 = ki0 % KT_DIM;
      int ki1 = ki0 + 1;
      int i1 = ki1 / KT_DIM, d1 = ki1 % KT_DIM;
#pragma unroll
      for (int nt = 0; nt < 5; ++nt) {
        int tl = wn * 80 + nt * 16 + lr;   // <= 319; +dt <= 327 < ZS_COLS
        v2f b;
        b.x = Zs[i0 * ZS_COLS + tl + d0];
        b.y = Zs[i1 * ZS_COLS + tl + d1];
        acc[0][nt] = wmma_f32(a[0], b, acc[0][nt]);
        acc[1][nt] = wmma_f32(a[1], b, acc[1][nt]);
      }
    }
  }

  // epilogue: +bias, store Y[n][o][t][w], accumulate BN statistics
#pragma unroll
  for (int mt = 0; mt < 2; ++mt) {
#pragma unroll
    for (int i = 0; i < 8; ++i) {
      int o = wm * 32 + mt * 16 + i + 8 * h;
      float bo = b_tcn[o];
      float ps = 0.f, pss = 0.f;
#pragma unroll
      for (int nt = 0; nt < 5; ++nt) {
        int t = wn * 80 + nt * 16 + lr;
        float val = acc[mt][nt][i] + bo;
        if (t < T_DIM) {
          Y[((size_t)(n * C_DIM + o) * T_DIM + t) * V_DIM + w] = val;
          ps  += val;
          pss += val * val;
        }
      }
      atomicAdd(&sstat[o], ps);
      atomicAdd(&sstat[64 + o], pss);
    }
  }
  __syncthreads();
  if (tid < 64) {
    atomicAdd(&gstat[tid], sstat[tid]);
    atomicAdd(&gstat[64 + tid], sstat[64 + tid]);
  }
}

// ---------------------------------------------------------------------------
// Kernel 3: finalize BN stats -> per-channel scale/shift
// ---------------------------------------------------------------------------
__global__ void finalize_stats(const float* __restrict__ gstat,
                               const float* __restrict__ gamma,
                               const float* __restrict__ beta,
                               float* __restrict__ scale,
                               float* __restrict__ shift) {
  int c = threadIdx.x;
  if (c >= C_DIM) return;
  const float inv_cnt = 1.f / (float)CNT_ELEMS;
  float mean = gstat[c] * inv_cnt;
  float var  = gstat[64 + c] * inv_cnt - mean * mean;
  float inv  = rsqrtf(var + 1e-5f);
  float sa   = gamma[c] * inv;
  scale[c] = sa;
  shift[c] = beta[c] - mean * sa;
}

// ---------------------------------------------------------------------------
// Kernel 4: out = relu(Y*scale[c] + shift[c] + x), float4 vectorized
//   (T*V = 7500 is a multiple of 4 -> each float4 stays within one channel)
// ---------------------------------------------------------------------------
__global__ __launch_bounds__(256) void bn_res_relu(
    const float* __restrict__ Y, const float* __restrict__ x,
    const float* __restrict__ scale, const float* __restrict__ shift,
    float* __restrict__ out) {
  const long total4 = (long)N_BATCH * C_DIM * T_DIM * V_DIM / 4;
  long idx4 = (long)blockIdx.x * 256 + threadIdx.x;
  if (idx4 >= total4) return;
  long base = idx4 * 4;
  int c = (int)((base / (T_DIM * V_DIM)) % C_DIM);
  float s = scale[c], b = shift[c];
  float4 y  = *(const float4*)(Y + base);
  float4 xv = *(const float4*)(x + base);
  float4 o;
  o.x = fmaxf(y.x * s + b + xv.x, 0.f);
  o.y = fmaxf(y.y * s + b + xv.y, 0.f);
  o.z = fmaxf(y.z * s + b + xv.z, 0.f);
  o.w = fmaxf(y.w * s + b + xv.w, 0.f);
  *(float4*)(out + base) = o;
}

// ---------------------------------------------------------------------------
// Host launcher
// ---------------------------------------------------------------------------
extern "C" void kernel_launch(void* const* d_in, const int* in_sizes, int n_in,
                              void* d_out, int out_size, void* d_ws, size_t ws_size,
                              hipStream_t stream) {
  const float* x      = (const float*)d_in[0];
  const float* A      = (const float*)d_in[1];
  const float* w_gcn  = (const float*)d_in[2];
  const float* b_gcn  = (const float*)d_in[3];
  const float* w_tcn  = (const float*)d_in[4];
  const float* b_tcn  = (const float*)d_in[5];
  const float* gamma  = (const float*)d_in[6];
  const float* beta   = (const float*)d_in[7];
  (void)in_sizes; (void)n_in; (void)out_size; (void)ws_size;

  float* ws    = (float*)d_ws;
  float* Mbig  = ws;                              // 1600*1600 = 2,560,000
  float* bterm = ws + 2560000;                    // 1600
  float* gstat = ws + 2561600;                    // 128 (sum, sumsq)
  float* scale = ws + 2561728;                    // 64
  float* shift = ws + 2561792;                    // 64
  float* Y     = ws + 2561856;                    // 30,720,000
  float* Z     = (float*)d_out;                   // reuse d_out as Z scratch

  // 0) fold GCN weight + adjacency into one matrix; bias term; zero stats
  build_mbig<<<(R_DIM * R_DIM + 255) / 256, 256, 0, stream>>>(w_gcn, A, Mbig);
  build_bterm<<<7, 256, 0, stream>>>(b_gcn, A, bterm, gstat);

  // 1) fused GCN GEMM -> Z[n, (c,w), t]
  gcn_gemm<<<dim3(5, 13, N_BATCH), 256, 0, stream>>>(x, Mbig, bterm, Z);

  // 2) temporal conv (implicit GEMM) + bias + BN statistics -> Y
  size_t smem = (size_t)(C_DIM * ZS_COLS + 16 * 64 + 128) * sizeof(float); // ~88.6 KB
  tcn_gemm<<<dim3(V_DIM, N_BATCH), 256, smem, stream>>>(Z, w_tcn, b_tcn, Y, gstat);

  // 3) finalize batch statistics
  finalize_stats<<<1, 64, 0, stream>>>(gstat, gamma, beta, scale, shift);

  // 4) BN affine + residual + ReLU -> d_out
  long total4 = (long)N_BATCH * C_DIM * T_DIM * V_DIM / 4;
  bn_res_relu<<<(int)((total4 + 255) / 256), 256, 0, stream>>>(Y, x, scale, shift, (float*)d_out);
}